// SubbandNN_19791209300213
// MI455X (gfx1250) — compile-verified
//
#include <hip/hip_runtime.h>

typedef __attribute__((ext_vector_type(2))) float v2f;
typedef __attribute__((ext_vector_type(8))) float v8f;
typedef __attribute__((ext_vector_type(4))) unsigned int u32x4;
typedef __attribute__((ext_vector_type(8))) int i32x8;
typedef __attribute__((ext_vector_type(4))) int i32x4;

#define B_N   4096
#define S_N   32
#define IN_N  256
#define H_N   128
#define OUT_N 256
#define BM    64          // rows of B per workgroup
#define XPAD  260         // 256 + 4 (4-bank skew per row -> conflict-free strided reads)
#define HPAD  132         // 128 + 4

#if defined(__gfx1250__) && __has_builtin(__builtin_amdgcn_tensor_load_to_lds)
#define HAVE_TDM_LD 1
#else
#define HAVE_TDM_LD 0
#endif
#if defined(__gfx1250__) && __has_builtin(__builtin_amdgcn_tensor_store_from_lds)
#define HAVE_TDM_ST 1
#else
#define HAVE_TDM_ST 0
#endif

// Fused per-subband MLP: out = relu(x@W1 + b1) @ W2 + b2
// One workgroup = (subband s, 64-row block of B). 8 waves of 32 (wave32).
// Wave mapping: each wave owns one 16-wide n-strip (two in GEMM2) and walks all four 16-row
// m-strips, so every W1/W2 fragment is fetched exactly once per workgroup.
__global__ __launch_bounds__(256) void subband_mlp_wmma(
    const float* __restrict__ x,  const float* __restrict__ W1,
    const float* __restrict__ b1, const float* __restrict__ W2,
    const float* __restrict__ b2, float* __restrict__ out)
{
    __shared__ float Xs[BM * XPAD];   // x tile [64 x 256] (padded rows); reused as
                                      // out tile [64 x 256] (dense) for the TDM store
    __shared__ float Hs[BM * HPAD];   // relu hidden [64 x 128] (padded rows)

    const int tid    = threadIdx.x;
    const int lane   = tid & 31;
    const int wave   = tid >> 5;
    const int lane15 = lane & 15;
    const int khi    = (lane >> 4) * 2;          // K offset for upper half-wave (A/B frags)
    const int mhalf  = 8 * (lane >> 4);          // M offset for upper half-wave (C/D frags)

    const int s  = blockIdx.x & (S_N - 1);
    const int m0 = (blockIdx.x >> 5) * BM;

    // ---------------- stage X tile [BM x IN] into LDS ----------------
#if HAVE_TDM_LD
    // Tensor Data Mover: one DMA for the 64x256 f32 tile, hardware LDS row
    // padding of 4 DWORDs per 256 DWORDs (-> row stride = XPAD floats).
    if (wave == 0) {
        const unsigned long long ga =
            (unsigned long long)(const void*)(x + ((long)m0 * S_N + s) * IN_N);
        const unsigned lds_off = (unsigned)(unsigned long long)(uintptr_t)&Xs[0];
        u32x4 g0 = { 1u,                                  // count=1
                     lds_off,
                     (unsigned)ga,
                     (unsigned)((ga >> 32) & 0x1FFFFFFu) | (2u << 30) };  // type=2
        i32x8 g1 = { (int)((2u << 16) | (1u << 20) | (7u << 22) | (3u << 25)),
                     (int)(256u << 16),      // tensor_dim0 = 256
                     (int)(64u  << 16),      // tensor_dim1 = 64
                     (int)(256u << 16),      // tile_dim0   = 256
                     64,                     // tile_dim1   = 64
                     8192,                   // tensor_dim0_stride = S_N*IN_N
                     0, 0 };
        i32x4 gz4 = { 0, 0, 0, 0 };
        i32x8 gz8 = { 0, 0, 0, 0, 0, 0, 0, 0 };
        __builtin_amdgcn_tensor_load_to_lds(g0, g1, gz4, gz4, gz8, 0);
        __builtin_amdgcn_s_wait_tensorcnt(0);
    }
#else
    {
        const float* xb = x + ((long)m0 * S_N + s) * IN_N;
        #pragma unroll
        for (int i = 0; i < 16; ++i) {
            int f   = tid + i * 256;      // float4 index 0..4095
            int row = f >> 4;             // 16 float4 per row
            int c4  = f & 15;
            const float4 v = *(const float4*)(xb + (long)row * (S_N * IN_N) + c4 * 4);
            *(float4*)&Xs[row * XPAD + c4 * 4] = v;
        }
    }
#endif
    // Warm cache for the W2 stream consumed in GEMM2 (global_prefetch_b8).
    __builtin_prefetch(W2 + (long)s * H_N * OUT_N + tid * 64, 0, 3);
    __builtin_prefetch(W2 + (long)s * H_N * OUT_N + 16384 + tid * 64, 0, 3);
    __syncthreads();

    // ---------------- GEMM1: Hs = relu(X @ W1 + b1) ----------------
    {
        const int nb = wave * 16;                          // 8 waves cover H=128
        const float* w1p = W1 + (long)s * IN_N * H_N;      // [k][n], stride H_N
        const float* bP  = w1p + (long)khi * H_N + nb + lane15;

        v8f acc[4] = {v8f{}, v8f{}, v8f{}, v8f{}};
        #pragma unroll 4
        for (int k0 = 0; k0 < IN_N; k0 += 4) {
            const float* bk = bP + (long)k0 * H_N;
            v2f bb = { bk[0], bk[H_N] };                   // B frag: loaded once,
            #pragma unroll                                 // reused by 4 m-strips
            for (int ms = 0; ms < 4; ++ms) {
                v2f a = *(const v2f*)&Xs[(ms * 16 + lane15) * XPAD + khi + k0];
                acc[ms] = __builtin_amdgcn_wmma_f32_16x16x4_f32(
                    false, a, false, bb, (short)0, acc[ms], false, false);
            }
        }
        const float bias = b1[s * H_N + nb + lane15];
        #pragma unroll
        for (int ms = 0; ms < 4; ++ms) {
            #pragma unroll
            for (int v = 0; v < 8; ++v) {
                float h = acc[ms][v] + bias;
                h = h > 0.f ? h : 0.f;
                Hs[(ms * 16 + mhalf + v) * HPAD + nb + lane15] = h;
            }
        }
    }
    __syncthreads();   // Hs ready; Xs dead from here on (reused as out tile)

    // ---------------- GEMM2: out = Hs @ W2 + b2 ----------------
    {
        const int nb0 = wave * 32;                         // 8 waves cover OUT=256
        const float* w2p = W2 + (long)s * H_N * OUT_N;     // [k][n], stride OUT_N
        const float* bP  = w2p + (long)khi * OUT_N + nb0 + lane15;

        v8f acc[2][4] = {};
        #pragma unroll 2
        for (int k0 = 0; k0 < H_N; k0 += 4) {
            const float* bk = bP + (long)k0 * OUT_N;
            v2f bb0 = { bk[0],  bk[OUT_N] };
            v2f bb1 = { bk[16], bk[OUT_N + 16] };
            #pragma unroll
            for (int ms = 0; ms < 4; ++ms) {
                v2f a = *(const v2f*)&Hs[(ms * 16 + lane15) * HPAD + khi + k0];
                acc[0][ms] = __builtin_amdgcn_wmma_f32_16x16x4_f32(
                    false, a, false, bb0, (short)0, acc[0][ms], false, false);
                acc[1][ms] = __builtin_amdgcn_wmma_f32_16x16x4_f32(
                    false, a, false, bb1, (short)0, acc[1][ms], false, false);
            }
        }

#if HAVE_TDM_LD && HAVE_TDM_ST
        // Stage the dense 64x256 out tile in LDS (aliasing Xs), then DMA it to
        // global with tensor_store_from_lds (row stride S_N*OUT_N).
        float* Os = Xs;
        #pragma unroll
        for (int t = 0; t < 2; ++t) {
            const int ncol = nb0 + t * 16 + lane15;
            const float bias = b2[s * OUT_N + ncol];
            #pragma unroll
            for (int ms = 0; ms < 4; ++ms) {
                #pragma unroll
                for (int v = 0; v < 8; ++v)
                    Os[(ms * 16 + mhalf + v) * OUT_N + ncol] = acc[t][ms][v] + bias;
            }
        }
        __syncthreads();   // all C-fragments in LDS before the DMA reads them
        if (wave == 0) {
            const unsigned long long ga =
                (unsigned long long)(void*)(out + ((long)m0 * S_N + s) * OUT_N);
            const unsigned lds_off = (unsigned)(unsigned long long)(uintptr_t)&Os[0];
            u32x4 g0 = { 1u,
                         lds_off,
                         (unsigned)ga,
                         (unsigned)((ga >> 32) & 0x1FFFFFFu) | (2u << 30) };
            i32x8 g1 = { (int)(2u << 16),        // data_size=4B, no padding on store
                         (int)(256u << 16),      // tensor_dim0 = 256
                         (int)(64u  << 16),      // tensor_dim1 = 64
                         (int)(256u << 16),      // tile_dim0   = 256
                         64,                     // tile_dim1   = 64
                         8192,                   // tensor_dim0_stride = S_N*OUT_N
                         0, 0 };
            i32x4 gz4 = { 0, 0, 0, 0 };
            i32x8 gz8 = { 0, 0, 0, 0, 0, 0, 0, 0 };
            __builtin_amdgcn_tensor_store_from_lds(g0, g1, gz4, gz4, gz8, 0);
            __builtin_amdgcn_s_wait_tensorcnt(0);
        }
#else
        #pragma unroll
        for (int t = 0; t < 2; ++t) {
            const int ncol = nb0 + t * 16 + lane15;
            const float bias = b2[s * OUT_N + ncol];
            #pragma unroll
            for (int ms = 0; ms < 4; ++ms) {
                #pragma unroll
                for (int v = 0; v < 8; ++v) {
                    const long brow = m0 + ms * 16 + mhalf + v;
                    out[(brow * S_N + s) * OUT_N + ncol] = acc[t][ms][v] + bias;
                }
            }
        }
#endif
    }
}

extern "C" void kernel_launch(void* const* d_in, const int* in_sizes, int n_in,
                              void* d_out, int out_size, void* d_ws, size_t ws_size,
                              hipStream_t stream) {
    const float* x  = (const float*)d_in[0];
    const float* W1 = (const float*)d_in[1];
    const float* b1 = (const float*)d_in[2];
    const float* W2 = (const float*)d_in[3];
    const float* b2 = (const float*)d_in[4];
    float* out = (float*)d_out;
    (void)in_sizes; (void)n_in; (void)out_size; (void)d_ws; (void)ws_size;

    dim3 grid((B_N / BM) * S_N);   // 64 row-blocks x 32 subbands = 2048 WGs
    dim3 block(256);               // 8 waves (wave32)
    subband_mlp_wmma<<<grid, block, 0, stream>>>(x, W1, b1, W2, b2, out);
}